// EntityClassify_1675037246053
// MI455X (gfx1250) — compile-verified
//
#include <hip/hip_runtime.h>

typedef __attribute__((ext_vector_type(2))) float v2f;
typedef __attribute__((ext_vector_type(8))) float v8f;

#define H_DIM 128
#define O_DIM 64

// ------------------------------------------------------------------
// utility: zero a float buffer
// ------------------------------------------------------------------
__global__ void zero_f32_kernel(float* __restrict__ p, long n) {
    long i = (long)blockIdx.x * blockDim.x + threadIdx.x;
    if (i < n) p[i] = 0.0f;
}

// ------------------------------------------------------------------
// per-relation in-degree count (float so we can reuse buffer as inv-deg)
// ------------------------------------------------------------------
__global__ void deg_count_kernel(const int* __restrict__ dst,
                                 float* __restrict__ deg,
                                 int E, int N, long total) {
    long i = (long)blockIdx.x * blockDim.x + threadIdx.x;
    if (i >= total) return;
    int r = (int)(i / E);
    int d = dst[i];
    atomicAdd(&deg[(long)r * N + d], 1.0f);
}

__global__ void deg_recip_kernel(float* __restrict__ deg, long n) {
    long i = (long)blockIdx.x * blockDim.x + threadIdx.x;
    if (i < n) deg[i] = 1.0f / fmaxf(deg[i], 1.0f);
}

// ------------------------------------------------------------------
// layer-1 scatter: one wave (32 lanes) per edge, lane covers 4 of 128 cols.
// message = embed[src] * inv_deg[r][dst], atomically accumulated into h1[dst].
// ------------------------------------------------------------------
__global__ void layer1_scatter_kernel(const float* __restrict__ embed,
                                      const int* __restrict__ src,
                                      const int* __restrict__ dst,
                                      const float* __restrict__ deg_inv,
                                      float* __restrict__ h1,
                                      int E, int N, long n_edges) {
    long gid  = (long)blockIdx.x * blockDim.x + threadIdx.x;
    long edge = gid >> 5;
    int  lane = (int)(gid & 31);
    if (edge >= n_edges) return;               // uniform per wave
    int r = (int)(edge / E);
    int s = src[edge];
    int d = dst[edge];
    float w = deg_inv[(long)r * N + d];
    const float4 v = *(const float4*)(embed + (long)s * H_DIM + lane * 4);
    float* o = h1 + (long)d * H_DIM + lane * 4;
    atomicAdd(o + 0, v.x * w);
    atomicAdd(o + 1, v.y * w);
    atomicAdd(o + 2, v.z * w);
    atomicAdd(o + 3, v.w * w);
}

// ------------------------------------------------------------------
// h1 = relu(h1 + bias1)
// ------------------------------------------------------------------
__global__ void relu_bias_kernel(float* __restrict__ h1,
                                 const float* __restrict__ bias1, long n) {
    long i = (long)blockIdx.x * blockDim.x + threadIdx.x;
    if (i < n) h1[i] = fmaxf(h1[i] + bias1[i & (H_DIM - 1)], 0.0f);
}

// ------------------------------------------------------------------
// out[n][o] = bias2[o]  (pre-seed accumulator with the final bias)
// ------------------------------------------------------------------
__global__ void init_out_kernel(float* __restrict__ out,
                                const float* __restrict__ bias2, long n) {
    long i = (long)blockIdx.x * blockDim.x + threadIdx.x;
    if (i < n) out[i] = bias2[i & (O_DIM - 1)];
}

// ------------------------------------------------------------------
// x = h1 @ W   via V_WMMA_F32_16X16X4_F32.
// Block = 256 threads = 8 waves; block covers 128 rows, wave w owns rows
// [base + 16w, base + 16w + 15] x all 64 output cols (four 16x16 tiles).
//
// W is staged into LDS in a K-PAIRED layout so that each lane's B fragment
// {W[k][col], W[k+1][col]} is 8 contiguous, 8-byte-aligned bytes:
//     Wlds[(k>>1)*2*O_DIM + col*2 + (k&1)]
// -> a single ds_load_b64 per fragment into an even VGPR pair, no repack movs.
//
// Fragment layouts per CDNA5 ISA 7.12.2 (32-bit, wave32):
//   A 16x4 : lanes 0-15 -> {K=0,K=1}, lanes 16-31 -> {K=2,K=3}, M = lane&15
//   B 4x16 : lanes 0-15 -> {K=0,K=1}, lanes 16-31 -> {K=2,K=3}, N = lane&15
//   C 16x16: VGPR v -> M = v + 8*(lane>=16), N = lane&15
// ------------------------------------------------------------------
__global__ void __launch_bounds__(256)
matmul_wmma_kernel(const float* __restrict__ h1,
                   const float* __restrict__ W,   // [H_DIM][O_DIM] row-major
                   float* __restrict__ x,         // [N][O_DIM]
                   int N) {
    __shared__ float Wlds[H_DIM * O_DIM];          // 32 KB, K-paired layout
    int tid = threadIdx.x;
    for (int i = tid; i < H_DIM * O_DIM; i += 256) {
        int k = i >> 6;            // i / O_DIM
        int o = i & (O_DIM - 1);   // i % O_DIM
        Wlds[(k >> 1) * (2 * O_DIM) + o * 2 + (k & 1)] = W[i];
    }
    __syncthreads();

    int wave    = tid >> 5;
    int lane    = tid & 31;
    int lhalf   = lane >> 4;       // 0: lanes 0-15, 1: lanes 16-31
    int l15     = lane & 15;
    int koff    = lhalf * 2;       // K offset for this lane's A/B fragment
    int rowBase = blockIdx.x * 128 + wave * 16;

    int  arow   = rowBase + l15;
    long arow_c = (long)min(arow, N - 1) * H_DIM;   // clamp: EXEC stays full

    // per-lane B fragment base: pair row (k0/2 + lhalf), column element l15
    // b64 address = (pairRow * 2*O_DIM + col*2) * 4 bytes
    const v2f* bbase = (const v2f*)Wlds + lhalf * O_DIM + l15; // + nb later

    v8f acc0 = {}; v8f acc1 = {}; v8f acc2 = {}; v8f acc3 = {};

    for (int k0 = 0; k0 < H_DIM; k0 += 4) {
        v2f a = *(const v2f*)(h1 + arow_c + k0 + koff);        // global b64

        const v2f* bp = bbase + (k0 >> 1) * O_DIM;             // pair row k0/2
        v2f b0 = bp[0];                                        // ds_load_b64
        v2f b1 = bp[16];
        v2f b2 = bp[32];
        v2f b3 = bp[48];

        acc0 = __builtin_amdgcn_wmma_f32_16x16x4_f32(false, a, false, b0,
                                                     (short)0, acc0, false, false);
        acc1 = __builtin_amdgcn_wmma_f32_16x16x4_f32(false, a, false, b1,
                                                     (short)0, acc1, false, false);
        acc2 = __builtin_amdgcn_wmma_f32_16x16x4_f32(false, a, false, b2,
                                                     (short)0, acc2, false, false);
        acc3 = __builtin_amdgcn_wmma_f32_16x16x4_f32(false, a, false, b3,
                                                     (short)0, acc3, false, false);
    }

    int crow = rowBase + 8 * lhalf;
    #pragma unroll
    for (int v = 0; v < 8; ++v) {
        int row = crow + v;
        if (row < N) {
            float* o = x + (long)row * O_DIM + l15;
            o[0]  = acc0[v];
            o[16] = acc1[v];
            o[32] = acc2[v];
            o[48] = acc3[v];
        }
    }
}

// ------------------------------------------------------------------
// layer-2 scatter for one relation: one wave per edge, lane covers 2 of 64
// cols. message = x[src] * inv_deg_r[dst], accumulated into out[dst].
// ------------------------------------------------------------------
__global__ void layer2_scatter_kernel(const float* __restrict__ x,
                                      const int* __restrict__ src,
                                      const int* __restrict__ dst,
                                      const float* __restrict__ deg_inv_r,
                                      float* __restrict__ out,
                                      long n_edges) {
    long gid  = (long)blockIdx.x * blockDim.x + threadIdx.x;
    long edge = gid >> 5;
    int  lane = (int)(gid & 31);
    if (edge >= n_edges) return;               // uniform per wave
    int s = src[edge];
    int d = dst[edge];
    float w = deg_inv_r[d];
    const float2 v = *(const float2*)(x + (long)s * O_DIM + lane * 2);
    float* o = out + (long)d * O_DIM + lane * 2;
    atomicAdd(o + 0, v.x * w);
    atomicAdd(o + 1, v.y * w);
}

// ------------------------------------------------------------------
extern "C" void kernel_launch(void* const* d_in, const int* in_sizes, int n_in,
                              void* d_out, int out_size, void* d_ws, size_t ws_size,
                              hipStream_t stream) {
    const float* embed    = (const float*)d_in[0];  // [N,H]
    const float* weight   = (const float*)d_in[1];  // [R,H,O]
    const float* bias1    = (const float*)d_in[2];  // [H]
    const float* bias2    = (const float*)d_in[3];  // [O]
    const int*   edge_src = (const int*)d_in[4];    // [R,E]
    const int*   edge_dst = (const int*)d_in[5];    // [R,E]

    const int H = in_sizes[2];                 // 128
    const int O = in_sizes[3];                 // 64
    const int N = in_sizes[0] / H;             // 100000
    const int R = in_sizes[1] / (H * O);       // 4
    const int E = in_sizes[4] / R;             // 160000

    float* out     = (float*)d_out;
    float* deg_inv = (float*)d_ws;                   // [R,N]
    float* h1      = deg_inv + (long)R * N;          // [N,H]
    float* x       = h1 + (long)N * H;               // [N,O]

    const int BT = 256;

    // ---- degrees -> reciprocal (shared by both layers) ----
    long rn = (long)R * N;
    zero_f32_kernel<<<(unsigned)((rn + BT - 1) / BT), BT, 0, stream>>>(deg_inv, rn);
    long re = (long)R * E;
    deg_count_kernel<<<(unsigned)((re + BT - 1) / BT), BT, 0, stream>>>(
        edge_dst, deg_inv, E, N, re);
    deg_recip_kernel<<<(unsigned)((rn + BT - 1) / BT), BT, 0, stream>>>(deg_inv, rn);

    // ---- layer 1: summed right-normalized aggregation + bias + relu ----
    long nh = (long)N * H;
    zero_f32_kernel<<<(unsigned)((nh + BT - 1) / BT), BT, 0, stream>>>(h1, nh);
    long l1_threads = re * 32;                 // one wave per edge
    layer1_scatter_kernel<<<(unsigned)((l1_threads + BT - 1) / BT), BT, 0, stream>>>(
        embed, edge_src, edge_dst, deg_inv, h1, E, N, re);
    relu_bias_kernel<<<(unsigned)((nh + BT - 1) / BT), BT, 0, stream>>>(h1, bias1, nh);

    // ---- layer 2: per-relation WMMA matmul then aggregation into out ----
    long no = (long)N * O;
    init_out_kernel<<<(unsigned)((no + BT - 1) / BT), BT, 0, stream>>>(out, bias2, no);

    unsigned mm_blocks = (unsigned)((N + 127) / 128);
    long l2_threads = (long)E * 32;
    for (int r = 0; r < R; ++r) {
        matmul_wmma_kernel<<<mm_blocks, BT, 0, stream>>>(
            h1, weight + (long)r * H * O, x, N);
        layer2_scatter_kernel<<<(unsigned)((l2_threads + BT - 1) / BT), BT, 0, stream>>>(
            x, edge_src + (long)r * E, edge_dst + (long)r * E,
            deg_inv + (long)r * N, out, (long)E);
    }
}